// Attention_talking_head_46583215292895
// MI455X (gfx1250) — compile-verified
//
#include <hip/hip_runtime.h>
#include <hip/hip_bf16.h>
#include <cstdint>

// MI455X / gfx1250, wave32. Fused talking-head attention + GFSA reaction.
// attn_final = (1-2*lam)*A + 3*lam*A@A  (H_CONST=4)
// out_head   = attn_final @ v = (1-2*lam)*(A@v) + 3*lam*A@(A@v)
// -> avoids the O(N^3) per-head A@A; everything is fp32 WMMA (16x16x4).

typedef float v2f __attribute__((ext_vector_type(2)));
typedef float v8f __attribute__((ext_vector_type(8)));
typedef float f4v __attribute__((ext_vector_type(4)));

#define B_   4
#define N_   1024
#define C_   768
#define H_   12
#define HD_  64

#define AS_STRIDE 17   // padded LDS strides: break bank conflicts
#define BS_STRIDE 65

// ---------------------------------------------------------------------------
// Wave32 WMMA f32 GEMM tile: block = 256 threads (8 waves).
// Block tile: 128 rows x 64 cols, K stepped by 16, LDS-staged A and B.
// Each wave owns a 16-row strip and 4 accumulators (16x16 each -> 64 cols).
// Fragment layouts per CDNA5 ISA 7.12.2:
//   A 16x4 : a[j] = A[lane&15][2*(lane>>4) + j]
//   B 4x16 : b[j] = B[2*(lane>>4) + j][lane&15]
//   C 16x16: c[i] = C[8*(lane>>4) + i][lane&15]
// B_IS_TRANSPOSED: global B stored as [Ncols x K] row-major (weight matrices,
// k-matrix for q@k^T); else stored as [K x Ncols] row-major (v, w1).
// ---------------------------------------------------------------------------
template <bool B_IS_TRANSPOSED>
__device__ __forceinline__ void gemm_tile(const float* __restrict__ Ag, int lda,
                                          const float* __restrict__ Bg, int ldb,
                                          int K, v8f acc[4]) {
  __shared__ float As[128 * AS_STRIDE];
  __shared__ float Bs[16 * BS_STRIDE];
  const int tid = threadIdx.x;
  const int w   = tid >> 5;   // wave id 0..7
  const int L   = tid & 31;   // lane
  const int lh  = L >> 4;     // lane half
  const int l16 = L & 15;

  for (int kb = 0; kb < K; kb += 16) {
    // Stage A tile (128 x 16): 2 x float4 per thread, coalesced.
#pragma unroll
    for (int rep = 0; rep < 2; ++rep) {
      const int idx = tid + rep * 256;
      const int row = idx >> 2;
      const int kq  = (idx & 3) << 2;
      const f4v a4 = *(const f4v*)(Ag + (size_t)row * lda + kb + kq);
      float* d = &As[row * AS_STRIDE + kq];
      d[0] = a4.x; d[1] = a4.y; d[2] = a4.z; d[3] = a4.w;
    }
    // Stage B tile as Bs[k][n] (16 x 64).
    if constexpr (B_IS_TRANSPOSED) {
      const int n  = tid >> 2;
      const int kq = (tid & 3) << 2;
      const f4v b4 = *(const f4v*)(Bg + (size_t)n * ldb + kb + kq);
      Bs[(kq + 0) * BS_STRIDE + n] = b4.x;
      Bs[(kq + 1) * BS_STRIDE + n] = b4.y;
      Bs[(kq + 2) * BS_STRIDE + n] = b4.z;
      Bs[(kq + 3) * BS_STRIDE + n] = b4.w;
    } else {
      const int kk = tid >> 4;
      const int nq = (tid & 15) << 2;
      const f4v b4 = *(const f4v*)(Bg + (size_t)(kb + kk) * ldb + nq);
      float* d = &Bs[kk * BS_STRIDE + nq];
      d[0] = b4.x; d[1] = b4.y; d[2] = b4.z; d[3] = b4.w;
    }
    // Prefetch next K-slab of A into cache (global_prefetch_b8).
    if (kb + 16 < K) {
      __builtin_prefetch(Ag + (size_t)(tid & 127) * lda + kb + 16, 0, 0);
    }
    __syncthreads();

#pragma unroll
    for (int kk = 0; kk < 16; kk += 4) {
      const int ka = kk + (lh << 1);
      v2f a;
      a.x = As[(w * 16 + l16) * AS_STRIDE + ka];
      a.y = As[(w * 16 + l16) * AS_STRIDE + ka + 1];
#pragma unroll
      for (int g = 0; g < 4; ++g) {
        v2f bf;
        bf.x = Bs[(ka)     * BS_STRIDE + g * 16 + l16];
        bf.y = Bs[(ka + 1) * BS_STRIDE + g * 16 + l16];
        acc[g] = __builtin_amdgcn_wmma_f32_16x16x4_f32(
            /*neg_a=*/false, a, /*neg_b=*/false, bf,
            /*c_mod=*/(short)0, acc[g], /*reuse_a=*/false, /*reuse_b=*/false);
      }
    }
    __syncthreads();
  }
}

__device__ __forceinline__ void zero_acc(v8f acc[4]) {
  const v8f z = {0.f, 0.f, 0.f, 0.f, 0.f, 0.f, 0.f, 0.f};
#pragma unroll
  for (int g = 0; g < 4; ++g) acc[g] = z;
}

// ---------------------------------------------------------------------------
// K1: qkv = x @ qkv_w^T, scattered to q[b,h,n,d] (x SCALE), k[..], v[..]
// ---------------------------------------------------------------------------
__global__ void __launch_bounds__(256)
k_qkv_gemm(const float* __restrict__ x, const float* __restrict__ qkvw,
           float* __restrict__ q, float* __restrict__ kmat, float* __restrict__ v) {
  v8f acc[4]; zero_acc(acc);
  const int m0 = blockIdx.y * 128;
  const int c0 = blockIdx.x * 64;
  gemm_tile<true>(x + (size_t)m0 * C_, C_, qkvw + (size_t)c0 * C_, C_, C_, acc);

  const int tid = threadIdx.x, w = tid >> 5, L = tid & 31, lh = L >> 4, l16 = L & 15;
#pragma unroll
  for (int g = 0; g < 4; ++g) {
    const int gc    = c0 + g * 16 + l16;          // 0..2303
    const int which = gc / C_;                    // 0=q 1=k 2=v
    const int cc    = gc - which * C_;
    const int h = cc >> 6, d = cc & 63;
    float* dst = (which == 0) ? q : ((which == 1) ? kmat : v);
    const float sc = (which == 0) ? 0.125f : 1.0f;  // SCALE = 64^-0.5
#pragma unroll
    for (int i = 0; i < 8; ++i) {
      const int gm = m0 + w * 16 + lh * 8 + i;    // 0..4095
      const int b = gm >> 10, n = gm & (N_ - 1);
      dst[(((size_t)b * H_ + h) * N_ + n) * HD_ + d] = acc[g][i] * sc;
    }
  }
}

// ---------------------------------------------------------------------------
// K2: raw scores S[bh] = q[bh] @ k[bh]^T   ([1024x64]x[64x1024])
// ---------------------------------------------------------------------------
__global__ void __launch_bounds__(256)
k_scores(const float* __restrict__ q, const float* __restrict__ kmat,
         float* __restrict__ S) {
  v8f acc[4]; zero_acc(acc);
  const int bh = blockIdx.z;
  const float* qb = q    + (size_t)bh * N_ * HD_;
  const float* kb = kmat + (size_t)bh * N_ * HD_;
  float* Sb = S + ((size_t)bh << 20);
  const int n0 = blockIdx.y * 128;
  const int m0 = blockIdx.x * 64;
  gemm_tile<true>(qb + (size_t)n0 * HD_, HD_, kb + (size_t)m0 * HD_, HD_, HD_, acc);

  const int tid = threadIdx.x, w = tid >> 5, L = tid & 31, lh = L >> 4, l16 = L & 15;
#pragma unroll
  for (int g = 0; g < 4; ++g)
#pragma unroll
    for (int i = 0; i < 8; ++i)
      Sb[(size_t)(n0 + w * 16 + lh * 8 + i) * N_ + (m0 + g * 16 + l16)] = acc[g][i];
}

// ---------------------------------------------------------------------------
// K3: per (b,n): pre-softmax head mix -> softmax over m -> post-softmax mix.
// All 12 head-rows live in LDS (48KB); softmaxed rows live in registers.
// Reads S (raw) from A-buffer, overwrites with final mixed probabilities.
// ---------------------------------------------------------------------------
__global__ void __launch_bounds__(256)
k_mix_softmax_mix(const float* __restrict__ Lw, const float* __restrict__ lb,
                  const float* __restrict__ Ww, const float* __restrict__ wb,
                  float* __restrict__ A) {
  __shared__ float raw[H_][N_];
  __shared__ float red[256];
  __shared__ float Ls[H_ * H_], Ws[H_ * H_], lbs[H_], wbs[H_];
  const int tid = threadIdx.x;
  const int n = blockIdx.x;
  const int b = blockIdx.y;
  if (tid < H_ * H_) { Ls[tid] = Lw[tid]; Ws[tid] = Ww[tid]; }
  if (tid < H_)      { lbs[tid] = lb[tid]; wbs[tid] = wb[tid]; }

  const size_t rowbase = (size_t)b * H_ * N_ * N_ + (size_t)n * N_;
#pragma unroll
  for (int h = 0; h < H_; ++h)
    for (int m = tid; m < N_; m += 256)
      raw[h][m] = A[rowbase + (size_t)h * N_ * N_ + m];
  __syncthreads();

  float p[H_][4];  // softmaxed rows, 4 columns per thread per head
#pragma unroll
  for (int g = 0; g < H_; ++g) {
    float vals[4];
    float lmax = -3.402823466e38f;
#pragma unroll
    for (int j = 0; j < 4; ++j) {
      const int m = tid + j * 256;
      float s = lbs[g];
#pragma unroll
      for (int h = 0; h < H_; ++h) s = fmaf(Ls[g * H_ + h], raw[h][m], s);
      vals[j] = s;
      lmax = fmaxf(lmax, s);
    }
    red[tid] = lmax; __syncthreads();
    for (int off = 128; off > 0; off >>= 1) {
      if (tid < off) red[tid] = fmaxf(red[tid], red[tid + off]);
      __syncthreads();
    }
    const float rmax = red[0]; __syncthreads();
    float lsum = 0.f;
#pragma unroll
    for (int j = 0; j < 4; ++j) { vals[j] = __expf(vals[j] - rmax); lsum += vals[j]; }
    red[tid] = lsum; __syncthreads();
    for (int off = 128; off > 0; off >>= 1) {
      if (tid < off) red[tid] += red[tid + off];
      __syncthreads();
    }
    const float inv = 1.0f / red[0]; __syncthreads();
#pragma unroll
    for (int j = 0; j < 4; ++j) p[g][j] = vals[j] * inv;
  }

#pragma unroll
  for (int g = 0; g < H_; ++g)
#pragma unroll
    for (int j = 0; j < 4; ++j) {
      const int m = tid + j * 256;
      float s = wbs[g];
#pragma unroll
      for (int h = 0; h < H_; ++h) s = fmaf(Ws[g * H_ + h], p[h][j], s);
      A[rowbase + (size_t)g * N_ * N_ + m] = s;
    }
}

// ---------------------------------------------------------------------------
// K4: w1[bh] = A[bh] @ v[bh]   ([1024x1024]x[1024x64])
// ---------------------------------------------------------------------------
__global__ void __launch_bounds__(256)
k_av(const float* __restrict__ A, const float* __restrict__ v,
     float* __restrict__ w1) {
  v8f acc[4]; zero_acc(acc);
  const int bh = blockIdx.y;
  const float* Ab = A + ((size_t)bh << 20);
  const float* vb = v + (size_t)bh * N_ * HD_;
  float* w1b = w1 + (size_t)bh * N_ * HD_;
  const int n0 = blockIdx.x * 128;
  gemm_tile<false>(Ab + (size_t)n0 * N_, N_, vb, HD_, N_, acc);

  const int tid = threadIdx.x, w = tid >> 5, L = tid & 31, lh = L >> 4, l16 = L & 15;
#pragma unroll
  for (int g = 0; g < 4; ++g)
#pragma unroll
    for (int i = 0; i < 8; ++i)
      w1b[(size_t)(n0 + w * 16 + lh * 8 + i) * HD_ + (g * 16 + l16)] = acc[g][i];
}

// ---------------------------------------------------------------------------
// K5: head output = (1-2*lam)*w1 + 3*lam*(A @ w1), merged into [B,N,C] layout
// ---------------------------------------------------------------------------
__global__ void __launch_bounds__(256)
k_aav_out(const float* __restrict__ A, const float* __restrict__ w1,
          const float* __restrict__ lamb, float* __restrict__ hout) {
  v8f acc[4]; zero_acc(acc);
  const int bh = blockIdx.y;
  const int b = bh / H_, h = bh - b * H_;
  const float* Ab  = A  + ((size_t)bh << 20);
  const float* w1b = w1 + (size_t)bh * N_ * HD_;
  const int n0 = blockIdx.x * 128;
  gemm_tile<false>(Ab + (size_t)n0 * N_, N_, w1b, HD_, N_, acc);

  const float lam = lamb[h];
  const float c1 = 1.0f - 2.0f * lam;
  const float c3 = 3.0f * lam;
  const int tid = threadIdx.x, w = tid >> 5, L = tid & 31, lh = L >> 4, l16 = L & 15;
#pragma unroll
  for (int g = 0; g < 4; ++g)
#pragma unroll
    for (int i = 0; i < 8; ++i) {
      const int gn = n0 + w * 16 + lh * 8 + i;
      const int d  = g * 16 + l16;
      hout[((size_t)b * N_ + gn) * C_ + h * HD_ + d] =
          fmaf(c1, w1b[(size_t)gn * HD_ + d], c3 * acc[g][i]);
    }
}

// ---------------------------------------------------------------------------
// K6: out = hout @ proj_out_w^T + proj_out_b
// ---------------------------------------------------------------------------
__global__ void __launch_bounds__(256)
k_proj_out(const float* __restrict__ hout, const float* __restrict__ pw,
           const float* __restrict__ pb, float* __restrict__ out) {
  v8f acc[4]; zero_acc(acc);
  const int m0 = blockIdx.y * 128;
  const int c0 = blockIdx.x * 64;
  gemm_tile<true>(hout + (size_t)m0 * C_, C_, pw + (size_t)c0 * C_, C_, C_, acc);

  const int tid = threadIdx.x, w = tid >> 5, L = tid & 31, lh = L >> 4, l16 = L & 15;
#pragma unroll
  for (int g = 0; g < 4; ++g) {
    const int gc = c0 + g * 16 + l16;
    const float bias = pb[gc];
#pragma unroll
    for (int i = 0; i < 8; ++i) {
      const int gm = m0 + w * 16 + lh * 8 + i;
      out[(size_t)gm * C_ + gc] = acc[g][i] + bias;
    }
  }
}

// ---------------------------------------------------------------------------
extern "C" void kernel_launch(void* const* d_in, const int* in_sizes, int n_in,
                              void* d_out, int out_size, void* d_ws, size_t ws_size,
                              hipStream_t stream) {
  const float* x          = (const float*)d_in[0];
  const float* qkv_w      = (const float*)d_in[1];
  const float* proj_l_w   = (const float*)d_in[2];
  const float* proj_l_b   = (const float*)d_in[3];
  const float* proj_w_w   = (const float*)d_in[4];
  const float* proj_w_b   = (const float*)d_in[5];
  const float* lamb       = (const float*)d_in[6];
  const float* proj_out_w = (const float*)d_in[7];
  const float* proj_out_b = (const float*)d_in[8];
  float* out = (float*)d_out;

  // Workspace layout (floats):
  //   q, k, v, w1 : [B,H,N,HD] = 3,145,728 each
  //   hout        : [B,N,C]    = 3,145,728
  //   A           : [B,H,N,N]  = 50,331,648
  float* ws = (float*)d_ws;
  const size_t HSZ = (size_t)B_ * H_ * N_ * HD_;  // 3,145,728
  float* q    = ws;
  float* kmat = ws + 1 * HSZ;
  float* v    = ws + 2 * HSZ;
  float* w1   = ws + 3 * HSZ;
  float* hout = ws + 4 * HSZ;
  float* A    = ws + 5 * HSZ;

  const dim3 blk(256);
  // K1: QKV projection  [4096x768]x[768x2304]
  k_qkv_gemm<<<dim3(36, 32), blk, 0, stream>>>(x, qkv_w, q, kmat, v);
  // K2: raw scores per (b,h)
  k_scores<<<dim3(16, 8, 48), blk, 0, stream>>>(q, kmat, A);
  // K3: head mix -> softmax -> head mix, per (b,n) row
  k_mix_softmax_mix<<<dim3(1024, 4), blk, 0, stream>>>(proj_l_w, proj_l_b,
                                                       proj_w_w, proj_w_b, A);
  // K4: w1 = A @ v
  k_av<<<dim3(8, 48), blk, 0, stream>>>(A, v, w1);
  // K5: head out = (1-2l)w1 + 3l*A@w1, merged heads
  k_aav_out<<<dim3(8, 48), blk, 0, stream>>>(A, w1, lamb, hout);
  // K6: output projection + bias
  k_proj_out<<<dim3(12, 32), blk, 0, stream>>>(hout, proj_out_w, proj_out_b, out);
}